// SimSiam_49452253446594
// MI455X (gfx1250) — compile-verified
//
#include <hip/hip_runtime.h>

#define NROW 8192
#define DD   256
#define HH   128

typedef __attribute__((ext_vector_type(2))) float v2f;
typedef __attribute__((ext_vector_type(8))) float v8f;

// D = A(16x4, f32) * B(4x16, f32) + C(16x16, f32), full fp32 on the WMMA pipe.
__device__ __forceinline__ v8f wmma_f32(v2f a, v2f b, v8f c) {
  return __builtin_amdgcn_wmma_f32_16x16x4_f32(
      /*neg_a=*/false, a, /*neg_b=*/false, b,
      /*c_mod=*/(short)0, c, /*reuse_a=*/false, /*reuse_b=*/false);
}

// ---------------------------------------------------------------------------
// Fused 2-layer MLP: P = relu(V@W1 + b1) @ W2 + b2
// One wave owns a 16-row tile; 8 waves/block -> 128 rows/block.
// A-fragment layout (16x4 f32): lane l, reg r -> M = l%16, K = 2*(l/16)+r
// D-fragment layout (16x16 f32): lane l, reg r -> N = l%16, M = r + 8*(l/16)
// ---------------------------------------------------------------------------
__global__ __launch_bounds__(256) void mlp_kernel(
    const float* __restrict__ V, const float* __restrict__ W1,
    const float* __restrict__ b1, const float* __restrict__ W2,
    const float* __restrict__ b2, float* __restrict__ P) {
  __shared__ float h_lds[8][16 * HH];  // 64 KB: per-wave 16x128 hidden tile
  const int wave = threadIdx.x >> 5;
  const int lane = threadIdx.x & 31;
  const int ln   = lane & 15;
  const int half = lane >> 4;
  const int row0 = (blockIdx.x * 8 + wave) * 16;
  float* h = h_lds[wave];

  // Stage 1: h = relu(V_tile @ W1 + b1)
  for (int nc = 0; nc < HH; nc += 16) {
    v8f acc = {};
    for (int k0 = 0; k0 < DD; k0 += 4) {
      const int k = k0 + 2 * half;
      v2f a = *(const v2f*)(V + (size_t)(row0 + ln) * DD + k);
      v2f b;
      b.x = W1[(size_t)k * HH + nc + ln];
      b.y = W1[(size_t)(k + 1) * HH + nc + ln];
      acc = wmma_f32(a, b, acc);
    }
    const float bias = b1[nc + ln];
#pragma unroll
    for (int r = 0; r < 8; ++r) {
      float v = acc[r] + bias;
      h[(r + 8 * half) * HH + nc + ln] = v > 0.f ? v : 0.f;
    }
  }
  __syncthreads();

  // Stage 2: P_tile = h @ W2 + b2
  for (int nc = 0; nc < DD; nc += 16) {
    v8f acc = {};
    for (int k0 = 0; k0 < HH; k0 += 4) {
      const int k = k0 + 2 * half;
      v2f a = *(const v2f*)(h + ln * HH + k);  // ds_load_b64
      v2f b;
      b.x = W2[(size_t)k * DD + nc + ln];
      b.y = W2[(size_t)(k + 1) * DD + nc + ln];
      acc = wmma_f32(a, b, acc);
    }
    const float bias = b2[nc + ln];
#pragma unroll
    for (int r = 0; r < 8; ++r)
      P[(size_t)(row0 + r + 8 * half) * DD + nc + ln] = acc[r] + bias;
  }
}

// ---------------------------------------------------------------------------
// Row L2-normalize: dst[row] = src[row] / max(||src[row]||, 1e-12)
// One wave per 256-wide row.
// ---------------------------------------------------------------------------
__global__ __launch_bounds__(256) void normalize_kernel(
    const float* __restrict__ src, float* __restrict__ dst) {
  const int wave = threadIdx.x >> 5, lane = threadIdx.x & 31;
  const size_t row = (size_t)blockIdx.x * 8 + wave;
  const float* s = src + row * DD;
  float vals[8];
  float ss = 0.f;
#pragma unroll
  for (int i = 0; i < 8; ++i) {
    vals[i] = s[lane + 32 * i];
    ss += vals[i] * vals[i];
  }
#pragma unroll
  for (int off = 16; off; off >>= 1) ss += __shfl_xor(ss, off, 32);
  const float inv = 1.0f / fmaxf(sqrtf(ss), 1e-12f);
  float* d = dst + row * DD;
#pragma unroll
  for (int i = 0; i < 8; ++i) d[lane + 32 * i] = vals[i] * inv;
}

// Per-row dot products (loss terms); deterministic (no float atomics).
__global__ __launch_bounds__(256) void rowdot_kernel(
    const float* __restrict__ A, const float* __restrict__ B,
    float* __restrict__ out) {
  const int wave = threadIdx.x >> 5, lane = threadIdx.x & 31;
  const size_t row = (size_t)blockIdx.x * 8 + wave;
  const float* a = A + row * DD;
  const float* b = B + row * DD;
  float s = 0.f;
#pragma unroll
  for (int i = 0; i < 8; ++i) s += a[lane + 32 * i] * b[lane + 32 * i];
#pragma unroll
  for (int off = 16; off; off >>= 1) s += __shfl_xor(s, off, 32);
  if (lane == 0) out[row] = s;
}

__global__ void init_kernel(unsigned long long* best) {
  best[blockIdx.x * blockDim.x + threadIdx.x] = 0ull;
}

// Pack (value, index) so that u64 atomicMax == "max value, ties -> lowest index"
// (matches jnp.argmax first-occurrence tie-break).
__device__ __forceinline__ unsigned long long packArg(float v, unsigned idx) {
  unsigned u = __float_as_uint(v);
  u = (u & 0x80000000u) ? ~u : (u | 0x80000000u);  // monotone float->u32 map
  return ((unsigned long long)u << 32) | (0xFFFFFFFFu - idx);
}

// ---------------------------------------------------------------------------
// Streaming similarity argmax: sim = A @ B^T (never materialized).
// One wave per 16x64 output strip: 4 accumulators reuse each A-fragment 4x
// (load:wmma drops from 2:1 to 1.25:1). The 4 B tiles are 16 KB apart, so
// they share one address register with 24-bit immediate offsets.
// Tile-local reductions, then one atomicMax per strip row / per column.
// ---------------------------------------------------------------------------
__global__ __launch_bounds__(256) void sim_kernel(
    const float* __restrict__ A, const float* __restrict__ B,
    unsigned long long* __restrict__ rowbest,
    unsigned long long* __restrict__ colbest) {
  const int wave = threadIdx.x >> 5;
  const int lane = threadIdx.x & 31;
  const int ln = lane & 15, half = lane >> 4;
  const int row0 = (blockIdx.y * 8 + wave) * 16;
  const int col0 = blockIdx.x * 64;

  const float* arow = A + (size_t)(row0 + ln) * DD;
  const float* brow = B + (size_t)(col0 + ln) * DD;  // B-frag[k][n] = B[n][k]
  v8f acc[4] = {{}, {}, {}, {}};
  for (int k0 = 0; k0 < DD; k0 += 4) {
    const int k = k0 + 2 * half;
    v2f a = *(const v2f*)(arow + k);
#pragma unroll
    for (int t = 0; t < 4; ++t) {
      v2f b = *(const v2f*)(brow + (size_t)(16 * t) * DD + k);
      acc[t] = wmma_f32(a, b, acc[t]);
    }
  }

  // Row argmax over this strip's 64 columns.
#pragma unroll
  for (int r = 0; r < 8; ++r) {
    float v = acc[0][r];
    int c = col0 + ln;
#pragma unroll
    for (int t = 1; t < 4; ++t) {  // ascending t => lowest column wins ties
      float vt = acc[t][r];
      if (vt > v) { v = vt; c = col0 + 16 * t + ln; }
    }
#pragma unroll
    for (int off = 1; off < 16; off <<= 1) {
      float v2 = __shfl_xor(v, off, 32);
      int c2 = __shfl_xor(c, off, 32);
      if (v2 > v || (v2 == v && c2 < c)) { v = v2; c = c2; }
    }
    if (ln == 0)
      atomicMax(&rowbest[row0 + r + 8 * half], packArg(v, (unsigned)c));
  }

  // Col argmax over the strip's 16 rows, per 16-wide tile.
#pragma unroll
  for (int t = 0; t < 4; ++t) {
    float v = acc[t][0];
    int rr = row0 + 8 * half;
#pragma unroll
    for (int r = 1; r < 8; ++r)
      if (acc[t][r] > v) { v = acc[t][r]; rr = row0 + r + 8 * half; }
    float v2 = __shfl_xor(v, 16, 32);
    int r2 = __shfl_xor(rr, 16, 32);
    if (v2 > v || (v2 == v && r2 < rr)) { v = v2; rr = r2; }
    if (half == 0)
      atomicMax(&colbest[col0 + 16 * t + ln], packArg(v, (unsigned)rr));
  }
}

// ---------------------------------------------------------------------------
// Final deterministic reduction -> [loss, acc_x, acc_y]
// acc = 0.5 * mean(argmax(sim, axis=0)==gold) + mean(argmax(sim, axis=1)==gold)
// ---------------------------------------------------------------------------
__global__ __launch_bounds__(256) void finalize_kernel(
    const unsigned long long* __restrict__ rb_x,
    const unsigned long long* __restrict__ cb_x,
    const unsigned long long* __restrict__ rb_y,
    const unsigned long long* __restrict__ cb_y,
    const float* __restrict__ dots_x, const float* __restrict__ dots_y,
    float* __restrict__ out) {
  __shared__ float red[6][256];
  const int t = threadIdx.x;
  float s[6] = {0.f, 0.f, 0.f, 0.f, 0.f, 0.f};
  for (int i = t; i < NROW; i += 256) {
    unsigned irx = 0xFFFFFFFFu - (unsigned)(rb_x[i] & 0xFFFFFFFFull);
    unsigned icx = 0xFFFFFFFFu - (unsigned)(cb_x[i] & 0xFFFFFFFFull);
    unsigned iry = 0xFFFFFFFFu - (unsigned)(rb_y[i] & 0xFFFFFFFFull);
    unsigned icy = 0xFFFFFFFFu - (unsigned)(cb_y[i] & 0xFFFFFFFFull);
    if (irx == (unsigned)i) s[0] += 1.f;  // rowmatch_x (axis=1)
    if (icx == (unsigned)i) s[1] += 1.f;  // colmatch_x (axis=0)
    if (iry == (unsigned)i) s[2] += 1.f;
    if (icy == (unsigned)i) s[3] += 1.f;
    s[4] += dots_x[i];
    s[5] += dots_y[i];
  }
#pragma unroll
  for (int j = 0; j < 6; ++j) red[j][t] = s[j];
  __syncthreads();
  for (int stride = 128; stride; stride >>= 1) {
    if (t < stride)
#pragma unroll
      for (int j = 0; j < 6; ++j) red[j][t] += red[j][t + stride];
    __syncthreads();
  }
  if (t == 0) {
    const float invN = 1.0f / (float)NROW;
    out[0] = -0.5f * (red[4][0] + red[5][0]) * invN;      // loss
    out[1] = 0.5f * red[1][0] * invN + red[0][0] * invN;  // acc_x
    out[2] = 0.5f * red[3][0] * invN + red[2][0] * invN;  // acc_y
  }
}

extern "C" void kernel_launch(void* const* d_in, const int* in_sizes, int n_in,
                              void* d_out, int out_size, void* d_ws,
                              size_t ws_size, hipStream_t stream) {
  const float* x  = (const float*)d_in[0];
  const float* y  = (const float*)d_in[1];
  const float* W1 = (const float*)d_in[2];
  const float* b1 = (const float*)d_in[3];
  const float* W2 = (const float*)d_in[4];
  const float* b2 = (const float*)d_in[5];

  // Workspace layout (~33.8 MB total)
  float* px = (float*)d_ws;                         // p_x, then p_x_n in place
  float* py = px + (size_t)NROW * DD;               // p_y, then p_y_n
  float* xn = py + (size_t)NROW * DD;               // x normalized
  float* yn = xn + (size_t)NROW * DD;               // y normalized
  unsigned long long* rb_x = (unsigned long long*)(yn + (size_t)NROW * DD);
  unsigned long long* cb_x = rb_x + NROW;
  unsigned long long* rb_y = cb_x + NROW;
  unsigned long long* cb_y = rb_y + NROW;
  float* dots_x = (float*)(cb_y + NROW);
  float* dots_y = dots_x + NROW;

  mlp_kernel<<<NROW / 128, 256, 0, stream>>>(x, W1, b1, W2, b2, px);
  mlp_kernel<<<NROW / 128, 256, 0, stream>>>(y, W1, b1, W2, b2, py);

  normalize_kernel<<<NROW / 8, 256, 0, stream>>>(px, px);
  normalize_kernel<<<NROW / 8, 256, 0, stream>>>(py, py);
  normalize_kernel<<<NROW / 8, 256, 0, stream>>>(x, xn);
  normalize_kernel<<<NROW / 8, 256, 0, stream>>>(y, yn);

  rowdot_kernel<<<NROW / 8, 256, 0, stream>>>(px, yn, dots_x);
  rowdot_kernel<<<NROW / 8, 256, 0, stream>>>(py, xn, dots_y);

  init_kernel<<<(4 * NROW) / 256, 256, 0, stream>>>(rb_x);  // all 4 best arrays

  dim3 simGrid(NROW / 64, NROW / 128);
  sim_kernel<<<simGrid, 256, 0, stream>>>(px, yn, rb_x, cb_x);
  sim_kernel<<<simGrid, 256, 0, stream>>>(py, xn, rb_y, cb_y);

  finalize_kernel<<<1, 256, 0, stream>>>(rb_x, cb_x, rb_y, cb_y, dots_x,
                                         dots_y, (float*)d_out);
}